// IntraGraphAttention_12043088298418
// MI455X (gfx1250) — compile-verified
//
#include <hip/hip_runtime.h>
#include <hip/hip_bf16.h>
#include <math.h>

#define IN_DIM   128
#define HEADS    2
#define OUT_DIM  64
#define CDIM     128            // HEADS*OUT_DIM
#define NEG_SLOPE 0.2f

typedef __attribute__((ext_vector_type(2))) float v2f;
typedef __attribute__((ext_vector_type(8))) float v8f;

// ---------------------------------------------------------------------------
// helpers
// ---------------------------------------------------------------------------
__device__ __forceinline__ float eluf(float v) {
    return v > 0.0f ? v : (__expf(v) - 1.0f);
}
__device__ __forceinline__ float leakyf(float v) {
    return v > 0.0f ? v : NEG_SLOPE * v;
}
// float atomic max via int/uint punning (maps to global_atomic_max_i32 / min_u32)
__device__ __forceinline__ void atomicMaxFloat(float* addr, float val) {
    if (val >= 0.0f)
        atomicMax((int*)addr, __float_as_int(val));
    else
        atomicMin((unsigned int*)addr, __float_as_uint(val));
}

// ---------------------------------------------------------------------------
// kernel 0: init out = bias, seg_max = -inf, denom = 0
// ---------------------------------------------------------------------------
__global__ void k_init(float* __restrict__ out, const float* __restrict__ bias,
                       float* __restrict__ seg, float* __restrict__ den, int n) {
    long long tid = (long long)blockIdx.x * blockDim.x + threadIdx.x;
    long long total = (long long)n * CDIM;
    if (tid < total) out[tid] = bias[tid & (CDIM - 1)];
    if (tid < 2LL * n) { seg[tid] = -INFINITY; den[tid] = 0.0f; }
}

// ---------------------------------------------------------------------------
// kernel 1: h = elu(x) @ W via V_WMMA_F32_16X16X4_F32
// block = 256 threads (8 waves). Wave w computes rows [row0, row0+16).
// Dynamic LDS: sW (64KB, K-pair interleaved) + per-wave sX[16*128] (64KB).
//
// sW layout: sW[(k>>1)*256 + col*2 + (k&1)] = W[k][col]
//   => a lane's B operand (W[k][col], W[k+1][col]) is one aligned b64 load.
// ---------------------------------------------------------------------------
__global__ void __launch_bounds__(256)
k_gemm(const float* __restrict__ x, const float* __restrict__ W,
       float* __restrict__ h, int n) {
    extern __shared__ float smem[];
    float* sW = smem;                  // 128*128 (interleaved pairs)
    float* sX = smem + IN_DIM * CDIM;  // 8 * 16 * 128

    const int tid  = threadIdx.x;
    const int wv   = tid >> 5;         // wave id 0..7
    const int lane = tid & 31;
    const int l15  = lane & 15;
    const int half = lane >> 4;        // 0 for lanes 0-15, 1 for lanes 16-31

    // --- stage W into LDS (pair-interleaved): 4096 float4 / 256 threads ---
    {
        const float4* Wv = (const float4*)W;
        #pragma unroll
        for (int i = 0; i < 16; ++i) {
            int idx = tid + i * 256;         // float4 index
            int r    = idx >> 5;             // row (32 float4 per row)
            int col0 = (idx & 31) * 4;
            float4 f = Wv[idx];
            float* dst = sW + (r >> 1) * 256 + col0 * 2 + (r & 1);
            dst[0] = f.x; dst[2] = f.y; dst[4] = f.z; dst[6] = f.w;
        }
    }

    // --- stage this wave's 16x128 x-tile with ELU applied ---
    const int row0 = blockIdx.x * 128 + wv * 16;
    float* sXw = sX + wv * 16 * IN_DIM;
    #pragma unroll 4
    for (int r = 0; r < 16; ++r) {
        int grow = row0 + r;
        float4 v = make_float4(0.f, 0.f, 0.f, 0.f);
        if (grow < n) v = *(const float4*)(x + (long long)grow * IN_DIM + 4 * lane);
        float* dst = sXw + r * IN_DIM + 4 * lane;
        dst[0] = eluf(v.x); dst[1] = eluf(v.y); dst[2] = eluf(v.z); dst[3] = eluf(v.w);
    }
    __syncthreads();

    // --- 256 WMMAs: 32 K-steps x 8 N-tiles of 16 ---
    v8f acc[8];
    #pragma unroll
    for (int nt = 0; nt < 8; ++nt) acc[nt] = (v8f){};

    const float* aBase = sXw + l15 * IN_DIM + half * 2;  // this lane's A row
    for (int kt = 0; kt < 32; ++kt) {
        // A tile 16x4: lanes0-15 -> K=4kt,4kt+1 ; lanes16-31 -> K=4kt+2,4kt+3
        v2f a = *(const v2f*)(aBase + 4 * kt);
        // kpair = (4kt + 2*half) >> 1
        const float* bBase = sW + (2 * kt + half) * 256 + l15 * 2;
        #pragma unroll
        for (int nt = 0; nt < 8; ++nt) {
            v2f b = *(const v2f*)(bBase + nt * 32);   // one ds_load_b64
            acc[nt] = __builtin_amdgcn_wmma_f32_16x16x4_f32(
                false, a, false, b, (short)0, acc[nt], false, false);
        }
    }

    // --- store: VGPR r, lanes0-15 -> row r, lanes16-31 -> row r+8 ---
    const int rbase = row0 + half * 8;
    #pragma unroll
    for (int nt = 0; nt < 8; ++nt) {
        #pragma unroll
        for (int r = 0; r < 8; ++r) {
            int grow = rbase + r;
            if (grow < n)
                h[(long long)grow * CDIM + nt * 16 + l15] = acc[nt][r];
        }
    }
}

// ---------------------------------------------------------------------------
// kernel 2: a_src[n,h] = h[n,h,:].att_src[h,:]  (one thread per node*head)
// ---------------------------------------------------------------------------
__global__ void k_dot(const float* __restrict__ h,
                      const float* __restrict__ att_src,
                      const float* __restrict__ att_dst,
                      float* __restrict__ a_src, float* __restrict__ a_dst, int n) {
    int tid = blockIdx.x * blockDim.x + threadIdx.x;
    if (tid >= 2 * n) return;
    int node = tid >> 1, hh = tid & 1;
    const float* hp = h + (long long)node * CDIM + hh * OUT_DIM;
    const float* as = att_src + hh * OUT_DIM;
    const float* ad = att_dst + hh * OUT_DIM;
    float s = 0.f, d = 0.f;
    #pragma unroll 8
    for (int i = 0; i < OUT_DIM; ++i) {
        float v = hp[i];
        s += v * as[i];
        d += v * ad[i];
    }
    a_src[tid] = s;
    a_dst[tid] = d;
}

// ---------------------------------------------------------------------------
// kernel 3: segment max over destination (edges + self loops)
// ---------------------------------------------------------------------------
__global__ void k_segmax(const int* __restrict__ src, const int* __restrict__ dst,
                         const float* __restrict__ a_src, const float* __restrict__ a_dst,
                         float* __restrict__ seg, int E, int n) {
    int tid = blockIdx.x * blockDim.x + threadIdx.x;
    if (tid >= E + n) return;
    int s, d;
    if (tid < E) { s = src[tid]; d = dst[tid]; } else { s = d = tid - E; }
    #pragma unroll
    for (int hh = 0; hh < HEADS; ++hh) {
        float l = leakyf(a_src[2 * s + hh] + a_dst[2 * d + hh]);
        atomicMaxFloat(&seg[2 * d + hh], l);
    }
}

// ---------------------------------------------------------------------------
// kernel 4: denom = segment sum of exp(logit - seg_max)
// ---------------------------------------------------------------------------
__global__ void k_denom(const int* __restrict__ src, const int* __restrict__ dst,
                        const float* __restrict__ a_src, const float* __restrict__ a_dst,
                        const float* __restrict__ seg, float* __restrict__ den,
                        int E, int n) {
    int tid = blockIdx.x * blockDim.x + threadIdx.x;
    if (tid >= E + n) return;
    int s, d;
    if (tid < E) { s = src[tid]; d = dst[tid]; } else { s = d = tid - E; }
    #pragma unroll
    for (int hh = 0; hh < HEADS; ++hh) {
        float l = leakyf(a_src[2 * s + hh] + a_dst[2 * d + hh]);
        atomicAdd(&den[2 * d + hh], __expf(l - seg[2 * d + hh]));
    }
}

// ---------------------------------------------------------------------------
// kernel 5: out[dst] += alpha * h[src]; one wave per edge, lane = 4 columns
// lanes 0-15 -> cols 0..63 (head 0), lanes 16-31 -> cols 64..127 (head 1)
// ---------------------------------------------------------------------------
__global__ void __launch_bounds__(256)
k_aggregate(const int* __restrict__ src, const int* __restrict__ dst,
            const float* __restrict__ a_src, const float* __restrict__ a_dst,
            const float* __restrict__ seg, const float* __restrict__ den,
            const float* __restrict__ h, float* __restrict__ out, int E, int n) {
    int eid  = blockIdx.x * (blockDim.x >> 5) + (threadIdx.x >> 5);
    int lane = threadIdx.x & 31;
    if (eid >= E + n) return;
    int s, d;
    if (eid < E) { s = src[eid]; d = dst[eid]; } else { s = d = eid - E; }

    int col = lane * 4;
    int hh  = lane >> 4;
    float l = leakyf(a_src[2 * s + hh] + a_dst[2 * d + hh]);
    float alpha = __expf(l - seg[2 * d + hh]) / (den[2 * d + hh] + 1e-16f);

    float4 v = *(const float4*)(h + (long long)s * CDIM + col);
    float* op = out + (long long)d * CDIM + col;
    atomicAdd(op + 0, alpha * v.x);
    atomicAdd(op + 1, alpha * v.y);
    atomicAdd(op + 2, alpha * v.z);
    atomicAdd(op + 3, alpha * v.w);
}

// ---------------------------------------------------------------------------
// launch
// ---------------------------------------------------------------------------
extern "C" void kernel_launch(void* const* d_in, const int* in_sizes, int n_in,
                              void* d_out, int out_size, void* d_ws, size_t ws_size,
                              hipStream_t stream) {
    const float* x        = (const float*)d_in[0];
    const int*   ei       = (const int*)d_in[1];   // [2,E] flat: first E = src, next E = dst
    const float* W        = (const float*)d_in[2];
    const float* att_src  = (const float*)d_in[3];
    const float* att_dst  = (const float*)d_in[4];
    const float* bias     = (const float*)d_in[5];
    float* out = (float*)d_out;

    const int n = in_sizes[0] / IN_DIM;   // 100000
    const int E = in_sizes[1] / 2;        // 1600000
    const int* src = ei;
    const int* dst = ei + E;

    // workspace layout
    float* h     = (float*)d_ws;                       // n*128
    float* a_src = h + (long long)n * CDIM;            // 2n
    float* a_dst = a_src + 2 * (long long)n;           // 2n
    float* seg   = a_dst + 2 * (long long)n;           // 2n
    float* den   = seg   + 2 * (long long)n;           // 2n

    const int B = 256;
    long long initN = (long long)n * CDIM;
    int gInit = (int)((initN + B - 1) / B);
    k_init<<<gInit, B, 0, stream>>>(out, bias, seg, den, n);

    int gGemm = (n + 127) / 128;
    size_t ldsBytes = (size_t)(IN_DIM * CDIM + 8 * 16 * IN_DIM) * sizeof(float); // 128KB
    k_gemm<<<gGemm, B, ldsBytes, stream>>>(x, W, h, n);

    int gDot = (2 * n + B - 1) / B;
    k_dot<<<gDot, B, 0, stream>>>(h, att_src, att_dst, a_src, a_dst, n);

    int gEdge = (E + n + B - 1) / B;
    k_segmax<<<gEdge, B, 0, stream>>>(src, dst, a_src, a_dst, seg, E, n);
    k_denom <<<gEdge, B, 0, stream>>>(src, dst, a_src, a_dst, seg, den, E, n);

    int wavesPerBlock = B / 32;
    int gAgg = (E + n + wavesPerBlock - 1) / wavesPerBlock;
    k_aggregate<<<gAgg, B, 0, stream>>>(src, dst, a_src, a_dst, seg, den, h, out, E, n);
}